// SymLoss_28475633173110
// MI455X (gfx1250) — compile-verified
//
#include <hip/hip_runtime.h>
#include <stdint.h>

// -----------------------------------------------------------------------------
// SymLoss on MI455X (gfx1250).
//  - Gather/L2-bound reduction: 2M (point,plane) pairs, 12B gathers into a
//    12MB cp table (L2-resident on 192MB L2); points streamed once (6MB).
//  - CDNA5 paths used:
//      * v_wmma_f32_16x16x4_f32 for the batched Gram (n_hat n_hat^T) term
//      * global_load_async_to_lds_b32 double-buffered point staging (ASYNCcnt)
//      * wave32 shuffle reductions
//  - This revision: cp gathers as single b96 loads (3x fewer VMEM issues in the
//    dominant loop); wave-uniform batch via readfirstlane -> scalar addressing.
// -----------------------------------------------------------------------------

typedef __attribute__((ext_vector_type(2))) float v2f;
typedef __attribute__((ext_vector_type(8))) float v8f;
typedef __attribute__((ext_vector_type(3))) float v3f;
typedef v3f v3f_a4 __attribute__((aligned(4)));  // 12B load, 4B alignment

#define NB    32      // batches
#define NC    4       // planes per batch
#define NPTS  16384   // points per batch
#define RES   32
#define MAIN_BLOCKS  512
#define MAIN_THREADS 256
#define ITERS 4       // (NB*NPTS) / (MAIN_BLOCKS*MAIN_THREADS)

// ws float layout: [0,512): plane params (nx,ny,nz,d) per (b,c)
//                  [512, 512+MAIN_BLOCKS): per-block partial sums

// ---------------------------------------------------------------------------
// Kernel 1: normalize planes, compute reg term via WMMA Gram matrices.
// Wave w computes the 16x16 Gram of normals 16w..16w+15 (batches 4w..4w+3)
// with a single V_WMMA_F32_16X16X4_F32 (K=3 padded with 0). Only the four
// 4x4 diagonal blocks are the per-batch Grams; off-blocks are masked out.
// ---------------------------------------------------------------------------
__global__ __launch_bounds__(256) void prep_reg(const float* __restrict__ yp,
                                                float* __restrict__ ws,
                                                float* __restrict__ out) {
  __shared__ float4 nh[128];
  __shared__ float  qbuf[256];
  __shared__ float  rbuf[32];
  const int t = threadIdx.x;

  if (t < 128) {  // one thread per (b,c)
    float nx = yp[t * 4 + 0], ny = yp[t * 4 + 1], nz = yp[t * 4 + 2];
    float d  = yp[t * 4 + 3];
    float inv = 1.0f / sqrtf(nx * nx + ny * ny + nz * nz);
    float4 v = make_float4(nx * inv, ny * inv, nz * inv, d);
    nh[t] = v;
    ((float4*)ws)[t] = v;  // plane params for the main kernel
  }
  __syncthreads();

  const int wave = t >> 5, lane = t & 31;
  const int m = lane & 15;
  float4 nv = nh[wave * 16 + m];

  // A (16x4 f32): lanes 0-15 hold {K0,K1}={nx,ny}; lanes 16-31 hold {K2,K3}={nz,0}.
  // B layout mirrors A with M<->N, so passing A for both operands yields Gram.
  v2f a;
  if (lane < 16) { a.x = nv.x; a.y = nv.y; }
  else           { a.x = nv.z; a.y = 0.0f; }
  v8f c = {0.f, 0.f, 0.f, 0.f, 0.f, 0.f, 0.f, 0.f};
  c = __builtin_amdgcn_wmma_f32_16x16x4_f32(false, a, false, a, (short)0, c,
                                            false, false);

  // D layout: lane L, vgpr r -> (row = r + 8*(L>>4), col = L&15).
  // Keep only same-batch 4x4 blocks; subtract identity on the diagonal.
  const int col = m, half = lane >> 4, colb = col >> 2;
  float q = 0.0f;
#pragma unroll
  for (int r = 0; r < 8; ++r) {
    int row = r + 8 * half;
    if ((row >> 2) == colb) {
      float v = c[r] - ((row == col) ? 1.0f : 0.0f);
      q += v * v;
    }
  }
  qbuf[t] = q;
  __syncthreads();

  if (t < 32) {  // one thread per batch: sum its 8 candidate lanes, sqrt
    int g = t >> 2, bi = t & 3;
    float s = 0.0f;
#pragma unroll
    for (int j = 0; j < 4; ++j) {
      s += qbuf[g * 32 + 4 * bi + j];
      s += qbuf[g * 32 + 16 + 4 * bi + j];
    }
    rbuf[t] = sqrtf(s);
  }
  __syncthreads();

  if (t == 0) {
    float acc = 0.0f;
    for (int b = 0; b < NB; ++b) acc += rbuf[b];
    out[0] = 25.0f * acc * (1.0f / (float)NB);  // REG_COEF * mean(reg)
  }
}

// ---------------------------------------------------------------------------
// Kernel 2: reflexion term. Each wave double-buffers 32-point (384B) tiles
// into its private LDS slice via async global->LDS copies, then evaluates all
// 4 planes per point (reflect -> voxel clamp -> b96 cp gather -> norm).
// ---------------------------------------------------------------------------
__global__ __launch_bounds__(MAIN_THREADS) void reflex_main(
    const float* __restrict__ points, const float* __restrict__ cp,
    const float* __restrict__ ws, float* __restrict__ partials) {
  __shared__ float sbuf[8][2][96];  // [wave][buffer][32 points * 3]
  __shared__ float wsum[8];

  const int t = threadIdx.x;
  const int wave = t >> 5, lane = t & 31;
  const int waveLinear = blockIdx.x * (MAIN_THREADS / 32) + wave;
  const int totalWaves = MAIN_BLOCKS * (MAIN_THREADS / 32);

  auto issue = [&](int it, int buf) {
    unsigned tile = (unsigned)(waveLinear + it * totalWaves);
    unsigned p0   = tile * 32u;                       // first point of tile
    unsigned gOff = (p0 * 3u + (unsigned)lane) * 4u;  // byte offset into points
    unsigned lds0 = (unsigned)(uintptr_t)&sbuf[wave][buf][lane];
    // 3 x 32 lanes x b32 = 384B per tile; offset applies to both sides.
    asm volatile("global_load_async_to_lds_b32 %0, %1, %2 offset:0"
                 :: "v"(lds0), "v"(gOff), "s"(points) : "memory");
    asm volatile("global_load_async_to_lds_b32 %0, %1, %2 offset:128"
                 :: "v"(lds0), "v"(gOff), "s"(points) : "memory");
    asm volatile("global_load_async_to_lds_b32 %0, %1, %2 offset:256"
                 :: "v"(lds0), "v"(gOff), "s"(points) : "memory");
  };

  issue(0, 0);
  float acc = 0.0f;

  for (int it = 0; it < ITERS; ++it) {
    const int buf = it & 1;
    if (it + 1 < ITERS) {
      issue(it + 1, (it + 1) & 1);  // prefetch next tile into other buffer
      asm volatile("s_wait_asynccnt 3" ::: "memory");  // current tile done
    } else {
      asm volatile("s_wait_asynccnt 0" ::: "memory");
    }

    const unsigned p0 = (unsigned)(waveLinear + it * totalWaves) * 32u;
    // batch is wave-uniform (tiles never straddle a batch): make it scalar.
    const int batch = __builtin_amdgcn_readfirstlane((int)(p0 >> 14));
    const float4* pp = (const float4*)ws + batch * NC;
    const float* cpb = cp + (size_t)batch * (RES * RES * RES * 3);

    const volatile float* vb = &sbuf[wave][buf][0];
    float px = vb[lane * 3 + 0];
    float py = vb[lane * 3 + 1];
    float pz = vb[lane * 3 + 2];

#pragma unroll
    for (int c = 0; c < NC; ++c) {
      float4 pl = pp[c];  // (nx, ny, nz, d)
      float dist = fmaf(px, pl.x, fmaf(py, pl.y, fmaf(pz, pl.z, pl.w)));
      float tm = -2.0f * dist;
      float rx = fmaf(tm, pl.x, px);
      float ry = fmaf(tm, pl.y, py);
      float rz = fmaf(tm, pl.z, pz);
      int vx = (int)fminf(fmaxf(floorf(rx * 32.0f), 0.0f), 31.0f);
      int vy = (int)fminf(fmaxf(floorf(ry * 32.0f), 0.0f), 31.0f);
      int vz = (int)fminf(fmaxf(floorf(rz * 32.0f), 0.0f), 31.0f);
      int off = (((vx << 5) + vy) << 5) + vz;
      // single 12B gather (global_load_b96) instead of 3x b32
      v3f cv = *(const v3f_a4*)(cpb + off * 3);
      float dx = rx - cv.x, dy = ry - cv.y, dz = rz - cv.z;
      acc += sqrtf(fmaf(dx, dx, fmaf(dy, dy, dz * dz)));
    }
  }

  // wave32 butterfly reduce, then per-block sum (deterministic)
  for (int off = 16; off > 0; off >>= 1) acc += __shfl_xor(acc, off, 32);
  if (lane == 0) wsum[wave] = acc;
  __syncthreads();
  if (t == 0) {
    float s = 0.0f;
    for (int w = 0; w < 8; ++w) s += wsum[w];
    partials[blockIdx.x] = s;
  }
}

// ---------------------------------------------------------------------------
// Kernel 3: deterministic tree reduction of the 512 block partials.
// ---------------------------------------------------------------------------
__global__ __launch_bounds__(256) void finalize(const float* __restrict__ partials,
                                                float* __restrict__ out) {
  __shared__ float red[256];
  const int t = threadIdx.x;
  red[t] = partials[t] + partials[t + 256];
  __syncthreads();
  for (int stride = 128; stride > 0; stride >>= 1) {
    if (t < stride) red[t] += red[t + stride];
    __syncthreads();
  }
  if (t == 0) out[0] = out[0] + red[0] * (1.0f / (float)NPTS);
}

extern "C" void kernel_launch(void* const* d_in, const int* in_sizes, int n_in,
                              void* d_out, int out_size, void* d_ws, size_t ws_size,
                              hipStream_t stream) {
  const float* y_pred = (const float*)d_in[0];  // (32,4,4)
  const float* points = (const float*)d_in[1];  // (32,16384,3)
  // d_in[2] = voxel_grid: unused by the loss
  const float* cp     = (const float*)d_in[3];  // (32,32,32,32,3)
  float* out = (float*)d_out;
  float* ws  = (float*)d_ws;
  float* partials = ws + 512;

  prep_reg<<<1, 256, 0, stream>>>(y_pred, ws, out);
  reflex_main<<<MAIN_BLOCKS, MAIN_THREADS, 0, stream>>>(points, cp, ws, partials);
  finalize<<<1, 256, 0, stream>>>(partials, out);
}